// PainnModel_53815940219227
// MI455X (gfx1250) — compile-verified
//
#include <hip/hip_runtime.h>
#include <math.h>

// ---------------- constants ----------------
#define NA      5000
#define NA16    5008          // NA padded to 16
#define NE      250000
#define Fdim    256
#define F3      768
#define NBrbf   20
#define NLAYER  3
#define CHUNK_E 10000         // 25 exact chunks of edges (mult of 16)
#define PI_F    3.14159265358979323846f
#define KCUT    (PI_F / 5.0f) // pi / CUTOFF

typedef __attribute__((ext_vector_type(2))) float v2f;
typedef __attribute__((ext_vector_type(8))) float v8f;

#define WMMA_F32(a, b, c) \
    __builtin_amdgcn_wmma_f32_16x16x4_f32(false, (a), false, (b), (short)0, (c), false, false)

__device__ __forceinline__ float silu_f(float x) {
    return x / (1.0f + __expf(-x));
}

// =====================================================================
// Weight pre-pack: PW stores rows (k, k+1) interleaved so a wave's B
// fragment {W[k][n], W[k+1][n]} is one aligned b64 load.
//   PW[((k>>1)*N + n)*2 + (k&1)] = W[k*N + n]
// =====================================================================
__global__ void pack_pairs(const float* __restrict__ W, float* __restrict__ PW,
                           int K, int N)
{
    const int i = blockIdx.x * blockDim.x + threadIdx.x;
    if (i >= K * N) return;
    const int k = i / N, n = i - k * N;
    PW[((size_t)(k >> 1) * N + n) * 2 + (k & 1)] = W[i];
}

// =====================================================================
// fp32 WMMA GEMM:  C[M][N] = act(A[M][K] @ W[K][N] + bias[N])
// Block = 128 threads (4 waves). Wave: 32 rows (2 M-tiles) x 64 cols
// (4 N-tiles) = 8 accumulators; B fragments reused across both M-tiles.
// Software-pipelined: k-step (i+1) loads issued before step (i) WMMAs.
// Requires: M mult of 16, N mult of 64, K mult of 4 (and even).
// A-layout (16x4 f32): lane l -> row = l%16, VGPR v -> k = 2*(l/16)+v
// B-layout (4x16):     lane l -> col = l%16, VGPR v -> k = 2*(l/16)+v
// C-layout (16x16):    lane l -> col = l%16, VGPR i -> row = i + 8*(l/16)
// =====================================================================
__global__ __launch_bounds__(128)
void gemm_wmma_f32(const float* __restrict__ A, const float* __restrict__ PW,
                   const float* __restrict__ bias, float* __restrict__ C,
                   int Mp, int K, int N, int act)
{
    const int lane = threadIdx.x & 31;
    const int wid  = threadIdx.x >> 5;
    const int mbase = (blockIdx.y * 4 + wid) * 32;
    if (mbase >= Mp) return;                        // wave-uniform
    const int nbase = blockIdx.x * 64;
    const int r = lane & 15;
    const int h = lane >> 4;

    const bool t1ok = (mbase + 16) < Mp;            // wave-uniform
    const int row0 = mbase + r;
    const int row1 = t1ok ? (mbase + 16 + r) : row0;

    const float* arow0 = A + (size_t)row0 * K;
    const float* arow1 = A + (size_t)row1 * K;
    const v2f* pb0 = (const v2f*)PW + (size_t)h * N + nbase + r;  // k=0 pair

    v8f acc[8] = {};   // [mt*4 + nt]

    // preload k = 0
    v2f a0 = *(const v2f*)(arow0 + 2 * h);
    v2f a1 = *(const v2f*)(arow1 + 2 * h);
    v2f b0 = pb0[0], b1 = pb0[16], b2 = pb0[32], b3 = pb0[48];

    int k = 0;
    while (true) {
        const int kn = k + 4;
        const bool more = kn < K;
        v2f na0 = a0, na1 = a1, nb0 = b0, nb1 = b1, nb2 = b2, nb3 = b3;
        if (more) {
            na0 = *(const v2f*)(arow0 + kn + 2 * h);
            na1 = *(const v2f*)(arow1 + kn + 2 * h);
            const v2f* pb = pb0 + (size_t)(kn >> 1) * N;
            nb0 = pb[0]; nb1 = pb[16]; nb2 = pb[32]; nb3 = pb[48];
        }
        acc[0] = WMMA_F32(a0, b0, acc[0]);
        acc[1] = WMMA_F32(a0, b1, acc[1]);
        acc[2] = WMMA_F32(a0, b2, acc[2]);
        acc[3] = WMMA_F32(a0, b3, acc[3]);
        acc[4] = WMMA_F32(a1, b0, acc[4]);
        acc[5] = WMMA_F32(a1, b1, acc[5]);
        acc[6] = WMMA_F32(a1, b2, acc[6]);
        acc[7] = WMMA_F32(a1, b3, acc[7]);
        if (!more) break;
        a0 = na0; a1 = na1; b0 = nb0; b1 = nb1; b2 = nb2; b3 = nb3;
        k = kn;
    }

    #pragma unroll
    for (int t = 0; t < 4; ++t) {
        const int n = nbase + t * 16 + r;
        const float bn = bias[n];
        #pragma unroll
        for (int i = 0; i < 8; ++i) {
            const int m = mbase + i + 8 * h;
            float x = acc[t][i] + bn;
            if (act) x = silu_f(x);
            C[(size_t)m * N + n] = x;
        }
    }
    if (t1ok) {
        #pragma unroll
        for (int t = 0; t < 4; ++t) {
            const int n = nbase + t * 16 + r;
            const float bn = bias[n];
            #pragma unroll
            for (int i = 0; i < 8; ++i) {
                const int m = mbase + 16 + i + 8 * h;
                float x = acc[4 + t][i] + bn;
                if (act) x = silu_f(x);
                C[(size_t)m * N + n] = x;
            }
        }
    }
}

// =====================================================================
// Filter GEMM (fused):  FO[m][n] = (rbf(dist[e]) @ Wf + bf)[n] * fcut(e)
//                                  * scalar_out[send[e]][n],  e = e0 + m
// A (M x 20) generated on the fly: A[m][k] = sin(dist*(k+1)*pi/5)/dist
// Wf pre-packed in pairs. N = 768, K = 20. 2 M-tiles per wave.
// =====================================================================
__global__ __launch_bounds__(128)
void filt_gemm_fused(const float* __restrict__ edge_dist,
                     const int*   __restrict__ edge,
                     const float* __restrict__ PWf,
                     const float* __restrict__ bf,
                     const float* __restrict__ SO,
                     float* __restrict__ FO,
                     int e0, int Mp)
{
    const int lane = threadIdx.x & 31;
    const int wid  = threadIdx.x >> 5;
    const int mbase = (blockIdx.y * 4 + wid) * 32;
    if (mbase >= Mp) return;                        // wave-uniform
    const int nbase = blockIdx.x * 64;
    const int r = lane & 15;
    const int h = lane >> 4;

    const bool t1ok = (mbase + 16) < Mp;
    const int row0 = mbase + r;
    const int row1 = t1ok ? (mbase + 16 + r) : row0;

    const float d0 = edge_dist[e0 + row0];
    const float d1 = edge_dist[e0 + row1];
    const float i0 = 1.0f / d0;
    const float i1 = 1.0f / d1;
    const v2f* pb0 = (const v2f*)PWf + (size_t)h * F3 + nbase + r;

    v8f acc[8] = {};
    #pragma unroll
    for (int k = 0; k < NBrbf; k += 4) {
        const int k0 = k + 2 * h;
        v2f a0, a1;
        a0.x = __sinf(d0 * (float)(k0 + 1) * KCUT) * i0;
        a0.y = __sinf(d0 * (float)(k0 + 2) * KCUT) * i0;
        a1.x = __sinf(d1 * (float)(k0 + 1) * KCUT) * i1;
        a1.y = __sinf(d1 * (float)(k0 + 2) * KCUT) * i1;
        const v2f* pb = pb0 + (size_t)(k >> 1) * F3;
        const v2f b0 = pb[0], b1 = pb[16], b2 = pb[32], b3 = pb[48];
        acc[0] = WMMA_F32(a0, b0, acc[0]);
        acc[1] = WMMA_F32(a0, b1, acc[1]);
        acc[2] = WMMA_F32(a0, b2, acc[2]);
        acc[3] = WMMA_F32(a0, b3, acc[3]);
        acc[4] = WMMA_F32(a1, b0, acc[4]);
        acc[5] = WMMA_F32(a1, b1, acc[5]);
        acc[6] = WMMA_F32(a1, b2, acc[6]);
        acc[7] = WMMA_F32(a1, b3, acc[7]);
    }

    #pragma unroll
    for (int mt = 0; mt < 2; ++mt) {
        if (mt == 1 && !t1ok) continue;             // wave-uniform
        #pragma unroll
        for (int t = 0; t < 4; ++t) {
            const int n = nbase + t * 16 + r;
            const float bn = bf[n];
            #pragma unroll
            for (int i = 0; i < 8; ++i) {
                const int m = mbase + mt * 16 + i + 8 * h;
                const int e = e0 + m;
                const float d  = edge_dist[e];
                const float fc = (d < 5.0f) ? 0.5f * (__cosf(d * KCUT) + 1.0f) : 0.0f;
                const int snd  = edge[2 * e + 1];
                const float so = SO[(size_t)snd * F3 + n];
                FO[(size_t)m * F3 + n] = (acc[mt * 4 + t][i] + bn) * fc * so;
            }
        }
    }
}

// =====================================================================
// Edge scatter: one wave per edge; lane owns 8 CONTIGUOUS channels
// (b128 gathers). dS/dV accumulated with f32 atomics; reads pre-update V.
// =====================================================================
__global__ __launch_bounds__(256)
void edge_scatter(const int*   __restrict__ edge,
                  const float* __restrict__ edge_diff,
                  const float* __restrict__ edge_dist,
                  const float* __restrict__ FO,
                  const float* __restrict__ V,
                  float* __restrict__ dS,
                  float* __restrict__ dV,
                  int e0, int Mc)
{
    const int lane = threadIdx.x & 31;
    const int wid  = threadIdx.x >> 5;
    const int ew   = blockIdx.x * 8 + wid;
    if (ew >= Mc) return;
    const int e   = e0 + ew;
    const int rcv = edge[2 * e + 0];
    const int snd = edge[2 * e + 1];
    const float invd = 1.0f / edge_dist[e];
    const float ux = edge_diff[3 * e + 0] * invd;
    const float uy = edge_diff[3 * e + 1] * invd;
    const float uz = edge_diff[3 * e + 2] * invd;

    const int f0 = lane * 8;                 // 8 contiguous channels
    const int q  = lane * 2;                 // float4 index within 256-ch row

    const float4* fo4 = (const float4*)(FO + (size_t)ew * F3);
    const float4* v0q = (const float4*)(V + (size_t)(0 * NA16 + snd) * Fdim);
    const float4* v1q = (const float4*)(V + (size_t)(1 * NA16 + snd) * Fdim);
    const float4* v2q = (const float4*)(V + (size_t)(2 * NA16 + snd) * Fdim);

    float4 t;
    float gv[8], ge[8], ms[8], n0[8], n1[8], n2[8];
    t = fo4[q];          gv[0]=t.x; gv[1]=t.y; gv[2]=t.z; gv[3]=t.w;
    t = fo4[q + 1];      gv[4]=t.x; gv[5]=t.y; gv[6]=t.z; gv[7]=t.w;
    t = fo4[64 + q];     ge[0]=t.x; ge[1]=t.y; ge[2]=t.z; ge[3]=t.w;
    t = fo4[64 + q + 1]; ge[4]=t.x; ge[5]=t.y; ge[6]=t.z; ge[7]=t.w;
    t = fo4[128 + q];    ms[0]=t.x; ms[1]=t.y; ms[2]=t.z; ms[3]=t.w;
    t = fo4[128 + q + 1];ms[4]=t.x; ms[5]=t.y; ms[6]=t.z; ms[7]=t.w;
    t = v0q[q];          n0[0]=t.x; n0[1]=t.y; n0[2]=t.z; n0[3]=t.w;
    t = v0q[q + 1];      n0[4]=t.x; n0[5]=t.y; n0[6]=t.z; n0[7]=t.w;
    t = v1q[q];          n1[0]=t.x; n1[1]=t.y; n1[2]=t.z; n1[3]=t.w;
    t = v1q[q + 1];      n1[4]=t.x; n1[5]=t.y; n1[6]=t.z; n1[7]=t.w;
    t = v2q[q];          n2[0]=t.x; n2[1]=t.y; n2[2]=t.z; n2[3]=t.w;
    t = v2q[q + 1];      n2[4]=t.x; n2[5]=t.y; n2[6]=t.z; n2[7]=t.w;

    float* ds  = dS + (size_t)rcv * Fdim + f0;
    float* dv0 = dV + (size_t)(0 * NA16 + rcv) * Fdim + f0;
    float* dv1 = dV + (size_t)(1 * NA16 + rcv) * Fdim + f0;
    float* dv2 = dV + (size_t)(2 * NA16 + rcv) * Fdim + f0;

    #pragma unroll
    for (int j = 0; j < 8; ++j) {
        atomicAdd(ds + j,  ms[j]);
        atomicAdd(dv0 + j, n0[j] * gv[j] + ux * ge[j]);
        atomicAdd(dv1 + j, n1[j] * gv[j] + uy * ge[j]);
        atomicAdd(dv2 + j, n2[j] * gv[j] + uz * ge[j]);
    }
}

// ---------------- elementwise kernels ----------------
__global__ void init_nodes(const int* __restrict__ Z, const float* __restrict__ embed,
                           float* __restrict__ S, float* __restrict__ V)
{
    const int i = blockIdx.x * blockDim.x + threadIdx.x;
    if (i < 3 * NA16 * Fdim) V[i] = 0.0f;
    if (i < NA16 * Fdim) {
        const int a = i >> 8, f = i & 255;
        S[i] = (a < NA) ? embed[(size_t)Z[a] * Fdim + f] : 0.0f;
    }
}

__global__ void zero_buf(float* __restrict__ p, int n)
{
    const int i = blockIdx.x * blockDim.x + threadIdx.x;
    if (i < n) p[i] = 0.0f;
}

__global__ void apply_delta(float* __restrict__ S, float* __restrict__ V,
                            const float* __restrict__ dS, const float* __restrict__ dV)
{
    const int i = blockIdx.x * blockDim.x + threadIdx.x;
    if (i >= NA * Fdim) return;
    const int a = i >> 8, f = i & 255;
    S[(size_t)a * Fdim + f] += dS[(size_t)a * Fdim + f];
    #pragma unroll
    for (int d = 0; d < 3; ++d) {
        const size_t idx = (size_t)(d * NA16 + a) * Fdim + f;
        V[idx] += dV[idx];
    }
}

__global__ void vn_cat(const float* __restrict__ Vv, const float* __restrict__ S,
                       float* __restrict__ CAT)
{
    const int i = blockIdx.x * blockDim.x + threadIdx.x;
    if (i >= NA * Fdim) return;
    const int a = i >> 8, f = i & 255;
    const float v0 = Vv[(size_t)(0 * NA16 + a) * Fdim + f];
    const float v1 = Vv[(size_t)(1 * NA16 + a) * Fdim + f];
    const float v2 = Vv[(size_t)(2 * NA16 + a) * Fdim + f];
    CAT[(size_t)a * 512 + f]       = sqrtf(v0 * v0 + v1 * v1 + v2 * v2);
    CAT[(size_t)a * 512 + 256 + f] = S[(size_t)a * Fdim + f];
}

__global__ void update_nodes(const float* __restrict__ MOUT,
                             const float* __restrict__ Uv, const float* __restrict__ Vv,
                             float* __restrict__ V, float* __restrict__ S)
{
    const int i = blockIdx.x * blockDim.x + threadIdx.x;
    if (i >= NA * Fdim) return;
    const int a = i >> 8, f = i & 255;
    const float avv = MOUT[(size_t)a * F3 + f];
    const float asv = MOUT[(size_t)a * F3 + Fdim + f];
    const float ass = MOUT[(size_t)a * F3 + 2 * Fdim + f];
    float dot = 0.0f;
    #pragma unroll
    for (int d = 0; d < 3; ++d) {
        const size_t idx = (size_t)(d * NA16 + a) * Fdim + f;
        const float u = Uv[idx];
        dot += u * Vv[idx];
        V[idx] += avv * u;
    }
    S[(size_t)a * Fdim + f] += asv * dot + ass;
}

__global__ void readout_reduce(const float* __restrict__ H, const float* __restrict__ w2,
                               const float* __restrict__ b2, float* __restrict__ out)
{
    const int a = blockIdx.x * blockDim.x + threadIdx.x;
    if (a >= NA) return;
    float s = b2[0];
    const float4* h4 = (const float4*)(H + (size_t)a * Fdim);
    const float4* w4 = (const float4*)w2;
    for (int f = 0; f < Fdim / 4; ++f) {
        const float4 hv = h4[f], wv = w4[f];
        s += hv.x * wv.x + hv.y * wv.y + hv.z * wv.z + hv.w * wv.w;
    }
    out[a] = s;
}

// =====================================================================
// Host launcher
// =====================================================================
extern "C" void kernel_launch(void* const* d_in, const int* in_sizes, int n_in,
                              void* d_out, int out_size, void* d_ws, size_t ws_size,
                              hipStream_t stream)
{
    const int*   Z            = (const int*)  d_in[0];
    const int*   edge         = (const int*)  d_in[1];
    const float* edge_diff    = (const float*)d_in[2];
    const float* edge_dist    = (const float*)d_in[3];
    const float* embed        = (const float*)d_in[4];
    const float* msg_w_filter = (const float*)d_in[5];
    const float* msg_b_filter = (const float*)d_in[6];
    const float* msg_w1       = (const float*)d_in[7];
    const float* msg_b1       = (const float*)d_in[8];
    const float* msg_w2       = (const float*)d_in[9];
    const float* msg_b2       = (const float*)d_in[10];
    const float* upd_wU       = (const float*)d_in[11];
    const float* upd_bU       = (const float*)d_in[12];
    const float* upd_wV       = (const float*)d_in[13];
    const float* upd_bV       = (const float*)d_in[14];
    const float* upd_w1       = (const float*)d_in[15];
    const float* upd_b1       = (const float*)d_in[16];
    const float* upd_w2       = (const float*)d_in[17];
    const float* upd_b2       = (const float*)d_in[18];
    const float* ro_w1        = (const float*)d_in[19];
    const float* ro_b1        = (const float*)d_in[20];
    const float* ro_w2        = (const float*)d_in[21];
    const float* ro_b2        = (const float*)d_in[22];
    float* out = (float*)d_out;

    // workspace partition (floats)
    float* S   = (float*)d_ws;                 // [NA16][256]
    float* V   = S   + (size_t)NA16 * Fdim;    // [3][NA16][256]
    float* SO  = V   + (size_t)3 * NA16 * Fdim;// [NA16][768] (also MLP out)
    float* H1  = SO  + (size_t)NA16 * F3;      // [NA16][256] (dS / MLP hidden / RO hidden)
    float* Uv  = H1  + (size_t)NA16 * Fdim;    // [3][NA16][256] (also dV)
    float* Vv  = Uv  + (size_t)3 * NA16 * Fdim;// [3][NA16][256]
    float* CAT = Vv  + (size_t)3 * NA16 * Fdim;// [NA16][512]
    float* FO  = CAT + (size_t)NA16 * 512;     // [CHUNK_E][768]
    float* PW  = FO  + (size_t)CHUNK_E * F3;   // [256*768] packed-weight buffer

    const int EW = 256;
    const int n_sv = 3 * NA16 * Fdim;
    const int n_s  = NA16 * Fdim;
    const int n_ew = NA * Fdim;

    #define PACK(Wp, Kp, Np) \
        pack_pairs<<<((Kp) * (Np) + EW - 1) / EW, EW, 0, stream>>>((Wp), PW, (Kp), (Np))
    #define GEMM(Ap, bp, Cp, Mp, Kp, Np, actp) \
        gemm_wmma_f32<<<dim3((Np) / 64, ((Mp) + 127) / 128), 128, 0, stream>>>((Ap), PW, (bp), (Cp), (Mp), (Kp), (Np), (actp))

    init_nodes<<<(n_sv + EW - 1) / EW, EW, 0, stream>>>(Z, embed, S, V);

    for (int l = 0; l < NLAYER; ++l) {
        const float* w1  = msg_w1 + (size_t)l * Fdim * Fdim;
        const float* b1  = msg_b1 + (size_t)l * Fdim;
        const float* w2  = msg_w2 + (size_t)l * Fdim * F3;
        const float* b2  = msg_b2 + (size_t)l * F3;
        const float* wf  = msg_w_filter + (size_t)l * NBrbf * F3;
        const float* bf  = msg_b_filter + (size_t)l * F3;

        // scalar_out = silu(S@w1+b1)@w2+b2
        PACK(w1, Fdim, Fdim);
        GEMM(S, b1, H1, NA16, Fdim, Fdim, 1);
        PACK(w2, Fdim, F3);
        GEMM(H1, b2, SO, NA16, Fdim, F3, 0);

        // zero deltas (reuse H1 as dS, Uv as dV)
        zero_buf<<<(n_s + EW - 1) / EW, EW, 0, stream>>>(H1, n_s);
        zero_buf<<<(n_sv + EW - 1) / EW, EW, 0, stream>>>(Uv, n_sv);

        // chunked edge phase
        PACK(wf, NBrbf, F3);
        for (int e0 = 0; e0 < NE; e0 += CHUNK_E) {
            const int Mc = (NE - e0 < CHUNK_E) ? (NE - e0) : CHUNK_E;
            filt_gemm_fused<<<dim3(F3 / 64, (Mc + 127) / 128), 128, 0, stream>>>(
                edge_dist, edge, PW, bf, SO, FO, e0, Mc);
            edge_scatter<<<(Mc + 7) / 8, 256, 0, stream>>>(
                edge, edge_diff, edge_dist, FO, V, H1, Uv, e0, Mc);
        }

        apply_delta<<<(n_ew + EW - 1) / EW, EW, 0, stream>>>(S, V, H1, Uv);

        // update block
        const float* wU  = upd_wU + (size_t)l * Fdim * Fdim;
        const float* bU  = upd_bU + (size_t)l * Fdim;
        const float* wVw = upd_wV + (size_t)l * Fdim * Fdim;
        const float* bV  = upd_bV + (size_t)l * Fdim;
        const float* uw1 = upd_w1 + (size_t)l * 2 * Fdim * Fdim;
        const float* ub1 = upd_b1 + (size_t)l * Fdim;
        const float* uw2 = upd_w2 + (size_t)l * Fdim * F3;
        const float* ub2 = upd_b2 + (size_t)l * F3;

        PACK(wU, Fdim, Fdim);
        GEMM(V, bU, Uv, 3 * NA16, Fdim, Fdim, 0);
        PACK(wVw, Fdim, Fdim);
        GEMM(V, bV, Vv, 3 * NA16, Fdim, Fdim, 0);

        vn_cat<<<(n_ew + EW - 1) / EW, EW, 0, stream>>>(Vv, S, CAT);

        PACK(uw1, 2 * Fdim, Fdim);
        GEMM(CAT, ub1, H1, NA16, 2 * Fdim, Fdim, 1);
        PACK(uw2, Fdim, F3);
        GEMM(H1, ub2, SO, NA16, Fdim, F3, 0);

        update_nodes<<<(n_ew + EW - 1) / EW, EW, 0, stream>>>(SO, Uv, Vv, V, S);
    }

    // readout
    PACK(ro_w1, Fdim, Fdim);
    GEMM(S, ro_b1, H1, NA16, Fdim, Fdim, 1);
    readout_reduce<<<(NA + EW - 1) / EW, EW, 0, stream>>>(H1, ro_w2, ro_b2, out);

    #undef PACK
    #undef GEMM
}